// _System_contractive_23648089932550
// MI455X (gfx1250) — compile-verified
//
#include <hip/hip_runtime.h>
#include <math.h>

// ---------------------------------------------------------------------------
// Dimensions (fixed by the reference)
// ---------------------------------------------------------------------------
constexpr int NX = 256, NQ = 256, NU = 64;
constexpr int MT = 64;          // batch rows per workgroup in the fused kernel
constexpr int SP = 260;         // padded LDS stride for 256-wide tiles (260*4 % 16 == 0, kills bank conflicts)
constexpr int SU = 68;          // padded LDS stride for 64-wide u tile
constexpr float EPSC = 1e-3f;

typedef __attribute__((ext_vector_type(2))) float v2f;
typedef __attribute__((ext_vector_type(8))) float v8f;

// Packed weight layout: Wp[k>>2][n][k&3]  (the 4 K-values of one WMMA
// B-fragment contiguous per column).  One k4-group = 1024 floats.
__device__ __forceinline__ size_t pk4(int k, int n) {
  return (size_t)((k >> 2) * 1024 + n * 4 + (k & 3));
}

// ---------------------------------------------------------------------------
// WMMA helpers: D = A(16x4) * B(4x16) + C, fp32 in/out (V_WMMA_F32_16X16X4_F32)
// 8-arg signature per probe: (neg_a, A, neg_b, B, c_mod, C, reuse_a, reuse_b)
// ---------------------------------------------------------------------------
__device__ __forceinline__ v8f wmma4(v2f a, v2f b, v8f c) {
  return __builtin_amdgcn_wmma_f32_16x16x4_f32(false, a, false, b, (short)0, c,
                                               false, false);
}

// A-fragment (16x4 f32) from an LDS tile, row-major with stride `ld` (even).
// ISA 7.12.2: lane L<16 holds M=L,K={k,k+1}; lane L>=16 holds M=L-16,K={k+2,k+3}
__device__ __forceinline__ v2f ldsA(const float* s, int rbase, int k, int lane, int ld) {
  int r  = rbase + (lane & 15);
  int kk = k + ((lane >> 4) << 1);
  return *(const v2f*)(s + r * ld + kk);    // ds_load_b64, 8B aligned
}

// B-fragment (4x16 f32) from a fragment-packed weight matrix: one aligned
// global_load_b64 per lane; lanes L and L+16 read complementary 8B halves of
// the same 16B group -> perfectly coalesced, zero over-fetch.
__device__ __forceinline__ v2f glbB(const float* __restrict__ Wp, int k, int col, int lane) {
  int n = col + (lane & 15);
  const float* p = Wp + (size_t)((k >> 2) * 1024 + n * 4 + ((lane >> 4) << 1));
  return *(const v2f*)p;
}

// ---------------------------------------------------------------------------
// Kernel 1:  C(n x n) = alpha * A A^T + eps*I        (A is n x k, row-major)
// Used for H = X X^T + eps I   and   P = 0.5 Pstar Pstar^T + eps I
// ---------------------------------------------------------------------------
__global__ __launch_bounds__(256) void syrk_eps(const float* __restrict__ A,
                                                float* __restrict__ C, int n,
                                                int kdim, float alpha, float eps) {
  __shared__ float As[16][17];
  __shared__ float Bs[16][17];
  const int tx = threadIdx.x & 15, ty = threadIdx.x >> 4;
  const int gi = blockIdx.y * 16 + ty;
  const int gj = blockIdx.x * 16 + tx;
  float acc = 0.f;
  for (int kb = 0; kb < kdim; kb += 16) {
    As[ty][tx] = A[(blockIdx.y * 16 + ty) * kdim + kb + tx];
    Bs[ty][tx] = A[(blockIdx.x * 16 + ty) * kdim + kb + tx];
    __syncthreads();
#pragma unroll
    for (int kk = 0; kk < 16; ++kk) acc = fmaf(As[ty][kk], Bs[tx][kk], acc);
    __syncthreads();
  }
  C[gi * n + gj] = alpha * acc + ((gi == gj) ? eps : 0.f);
}

// ---------------------------------------------------------------------------
// Kernel 2: derived matrices (elementwise over 256x256)
//   Y    = -0.5 (H1 + Y1 - Y1^T)               (ALPHA == 0)
//   Z    = -H2 - Chi
//   C1p  : packed C1^T   [x][q] = lam_inv[q] * Chi[x][q]
//   D11p : packed D11^T  [j][i] = (j<i) ? -lam_inv[i]*H4[i][j] : 0
//   D12p, B2p : packed transposes (64 x 256)
// Row index i is the K-dimension of the downstream GEMMs -> pack on i.
// ---------------------------------------------------------------------------
__global__ __launch_bounds__(256) void derive(
    const float* __restrict__ H, const float* __restrict__ Chi,
    const float* __restrict__ Y1, const float* __restrict__ D12,
    const float* __restrict__ B2, float* __restrict__ Y, float* __restrict__ Z,
    float* __restrict__ D11p, float* __restrict__ C1p,
    float* __restrict__ D12p, float* __restrict__ B2p) {
  const int tx = threadIdx.x & 15, ty = threadIdx.x >> 4;
  const int i = blockIdx.y * 16 + ty;   // row (K index for packed outputs)
  const int j = blockIdx.x * 16 + tx;   // col
  const float lam_inv_j = 1.0f / (0.5f * H[(256 + j) * 512 + (256 + j)]);
  Y[i * 256 + j] = -0.5f * (H[i * 512 + j] + Y1[i * 256 + j] - Y1[j * 256 + i]);
  Z[i * 256 + j] = -H[i * 512 + 256 + j] - Chi[i * 256 + j];
  C1p[pk4(i, j)] = lam_inv_j * Chi[i * 256 + j];
  // D11^T[i][j] = D11[j][i] = -lam_inv[j] * H4[j][i]  for i<j (strict lower of D11)
  D11p[pk4(i, j)] =
      (i < j) ? (-lam_inv_j * H[(256 + j) * 512 + (256 + i)]) : 0.f;
  if (i < 64) {
    D12p[pk4(i, j)] = D12[j * 64 + i];
    B2p[pk4(i, j)] = B2[j * 64 + i];
  }
}

// ---------------------------------------------------------------------------
// Kernel 3: in-place Gauss-Jordan inverse of SPD 256x256, entirely in LDS.
// One workgroup, 1024 threads: thread (j = tid&255, q = tid>>8) owns
// column j, rows [64q, 64q+64). 257-float row stride avoids bank conflicts.
// Needs ~258 KB dynamic LDS (legal: CDNA5 allows 320 KB per workgroup).
// ---------------------------------------------------------------------------
__global__ __launch_bounds__(1024, 1) void gj_inverse(const float* __restrict__ P,
                                                      float* __restrict__ Pinv) {
  extern __shared__ float sm[];
  float* A = sm;                 // 256 * 257
  float* facs = A + 256 * 257;   // 256
  const int tid = threadIdx.x;
  const int j = tid & 255;
  const int q = tid >> 8;
  for (int r = 0; r < 64; ++r) {
    int i = q * 64 + r;
    A[i * 257 + j] = P[i * 256 + j];
  }
  __syncthreads();
  for (int k = 0; k < 256; ++k) {
    if (tid < 256) facs[tid] = A[tid * 257 + k];   // capture column k (pre-update)
    const float pinv = 1.0f / A[k * 257 + k];
    __syncthreads();
    if (q == 0 && j != k) A[k * 257 + j] *= pinv;  // scale pivot row
    __syncthreads();
    const float rkj = A[k * 257 + j];
    if (j != k) {
      for (int r = 0; r < 64; ++r) {
        int i = q * 64 + r;
        if (i != k) A[i * 257 + j] = fmaf(-facs[i], rkj, A[i * 257 + j]);
      }
    } else {
      for (int r = 0; r < 64; ++r) {
        int i = q * 64 + r;
        if (i != k) A[i * 257 + j] = -facs[i] * pinv;
      }
      if (q == 0) A[k * 257 + k] = pinv;
    }
    __syncthreads();
  }
  for (int r = 0; r < 64; ++r) {
    int i = q * 64 + r;
    Pinv[i * 256 + j] = A[i * 257 + j];
  }
}

// ---------------------------------------------------------------------------
// Kernel 4: Cp = pack4(A^T * B)   (256x256x256), used for
//   ATp  = pack(Y^T Pinv)   (A^T, since Pinv symmetric)
//   B1p  = pack(Z^T Pinv)   (B1^T)
// Output row gi is the K-dimension of the downstream GEMM -> pack on gi.
// ---------------------------------------------------------------------------
__global__ __launch_bounds__(256) void gemm_tn(const float* __restrict__ A,
                                               const float* __restrict__ B,
                                               float* __restrict__ Cp) {
  __shared__ float As[16][17];
  __shared__ float Bs[16][17];
  const int tx = threadIdx.x & 15, ty = threadIdx.x >> 4;
  const int gi = blockIdx.y * 16 + ty;
  const int gj = blockIdx.x * 16 + tx;
  float acc = 0.f;
  for (int kb = 0; kb < 256; kb += 16) {
    As[ty][tx] = A[(kb + ty) * 256 + blockIdx.y * 16 + tx];
    Bs[ty][tx] = B[(kb + ty) * 256 + blockIdx.x * 16 + tx];
    __syncthreads();
#pragma unroll
    for (int kk = 0; kk < 16; ++kk) acc = fmaf(As[kk][ty], Bs[kk][tx], acc);
    __syncthreads();
  }
  Cp[pk4(gi, gj)] = acc;
}

// ---------------------------------------------------------------------------
// Kernel 5: fused batch kernel. 512 blocks x 256 threads (8 wave32s).
// Block owns 64 batch rows; wave wv owns row-block (wv&3)*16 and a 128-wide
// column half (wv>>2). All GEMM work is V_WMMA_F32_16X16X4_F32.
// Phases: stage -> v_base GEMM -> 16-wide blocked tanh recurrence -> out GEMM.
// ---------------------------------------------------------------------------
__global__ __launch_bounds__(256, 1) void rren_main(
    const float* __restrict__ xi, const float* __restrict__ u,
    const float* __restrict__ C1p, const float* __restrict__ D12p,
    const float* __restrict__ D11p, const float* __restrict__ ATp,
    const float* __restrict__ B1p, const float* __restrict__ B2p,
    const float* __restrict__ bx, const float* __restrict__ bv,
    float* __restrict__ out) {
  extern __shared__ float smem[];
  float* xi_s = smem;                 // MT * SP
  float* w_s  = xi_s + MT * SP;       // MT * SP
  float* vb_s = w_s + MT * SP;        // MT * SP
  float* u_s  = vb_s + MT * SP;       // MT * SU
  float* d11b = u_s + MT * SU;        // 16 * 17
  float* red_s = d11b + 16 * 17;      // MT * 16 (deterministic split-K reduce)

  const int tid = threadIdx.x;
  const int lane = tid & 31;
  const int wv = tid >> 5;
  const int r0 = blockIdx.x * MT;
  const int rv = (wv & 3) * 16;       // row block inside tile
  const int c0 = (wv >> 2) * 128;     // column half

  // L2 prefetch hints for the hot weight streams (global_prefetch_b8)
  __builtin_prefetch(C1p + ((size_t)tid << 6), 0, 1);
  __builtin_prefetch(D11p + ((size_t)tid << 6), 0, 1);

  // ---- Phase 1: stage xi (64x256) and u (64x64) tiles into LDS -------------
  for (int e = tid * 4; e < MT * NX; e += 256 * 4) {
    const int row = e >> 8, col = e & 255;
    *(float4*)(xi_s + row * SP + col) =
        *(const float4*)(xi + (size_t)(r0 + row) * NX + col);
  }
  for (int e = tid * 4; e < MT * NU; e += 256 * 4) {
    const int row = e >> 6, col = e & 63;
    *(float4*)(u_s + row * SU + col) =
        *(const float4*)(u + (size_t)(r0 + row) * NU + col);
  }
  __syncthreads();

  const v8f vzero = {0.f, 0.f, 0.f, 0.f, 0.f, 0.f, 0.f, 0.f};

  // ---- Phase 2: v_base = xi @ C1^T + u @ D12^T + bv ------------------------
  {
    v8f acc[8];
#pragma unroll
    for (int t = 0; t < 8; ++t) acc[t] = vzero;
    for (int k = 0; k < NX; k += 4) {
      const v2f a = ldsA(xi_s, rv, k, lane, SP);
#pragma unroll
      for (int t = 0; t < 8; ++t)
        acc[t] = wmma4(a, glbB(C1p, k, c0 + 16 * t, lane), acc[t]);
    }
    for (int k = 0; k < NU; k += 4) {
      const v2f a = ldsA(u_s, rv, k, lane, SU);
#pragma unroll
      for (int t = 0; t < 8; ++t)
        acc[t] = wmma4(a, glbB(D12p, k, c0 + 16 * t, lane), acc[t]);
    }
#pragma unroll
    for (int t = 0; t < 8; ++t) {
      const int col = c0 + 16 * t + (lane & 15);
      const float bvv = bv[col];
#pragma unroll
      for (int v = 0; v < 8; ++v) {
        const int row = rv + v + ((lane >> 4) << 3);
        vb_s[row * SP + col] = acc[t][v] + bvv;
      }
    }
  }
  __syncthreads();

  // ---- Phase 3: blocked triangular tanh recurrence over 16 column blocks ---
  for (int kb = 0; kb < 16; ++kb) {
    const int K = kb * 16;             // columns [K, K+16) this step
    const int kh = wv >> 2;            // split-K half
    v8f pacc = vzero;
    if (K > 0) {
      const int klen = K >> 1;         // 8*kb, always a multiple of 4
      const int ks = kh * klen;
      const int ke = ks + klen;
      for (int k = ks; k < ke; k += 4) {
        const v2f a = ldsA(w_s, rv, k, lane, SP);
        pacc = wmma4(a, glbB(D11p, k, K, lane), pacc);
      }
      if (kh == 1) {                   // stage partial for deterministic reduce
        const int c = lane & 15;
#pragma unroll
        for (int v = 0; v < 8; ++v) {
          const int row = rv + v + ((lane >> 4) << 3);
          red_s[row * 16 + c] = pacc[v];
        }
      }
    }
    {  // stage D11 block: d11b[i][j] = D11[K+i][K+j] = D11^T[(K+j)][(K+i)]
      const int i = tid >> 4, jj = tid & 15;
      d11b[i * 17 + jj] = D11p[pk4(K + jj, K + i)];
    }
    __syncthreads();
    if (K > 0 && kh == 0) {            // deterministic accumulate into vb_s
      const int col = K + (lane & 15);
#pragma unroll
      for (int v = 0; v < 8; ++v) {
        const int row = rv + v + ((lane >> 4) << 3);
        vb_s[row * SP + col] += pacc[v] + red_s[row * 16 + (lane & 15)];
      }
    }
    __syncthreads();
    // serial intra-block chain: one thread per batch row, registers only
    if (tid < MT) {
      float wloc[16];
      const float* vr = vb_s + tid * SP + K;
#pragma unroll
      for (int i = 0; i < 16; ++i) {
        float v = vr[i];
#pragma unroll
        for (int j = 0; j < 16; ++j)
          if (j < i) v = fmaf(wloc[j], d11b[i * 17 + j], v);
        wloc[i] = tanhf(v);
      }
#pragma unroll
      for (int i = 0; i < 16; ++i) w_s[tid * SP + K + i] = wloc[i];
    }
    __syncthreads();
  }

  // ---- Phase 4: out = xi @ A^T + w @ B1^T + u @ B2^T + bx ------------------
  {
    v8f acc[8];
#pragma unroll
    for (int t = 0; t < 8; ++t) acc[t] = vzero;
    for (int k = 0; k < NX; k += 4) {
      const v2f a = ldsA(xi_s, rv, k, lane, SP);
#pragma unroll
      for (int t = 0; t < 8; ++t)
        acc[t] = wmma4(a, glbB(ATp, k, c0 + 16 * t, lane), acc[t]);
    }
    for (int k = 0; k < NQ; k += 4) {
      const v2f a = ldsA(w_s, rv, k, lane, SP);
#pragma unroll
      for (int t = 0; t < 8; ++t)
        acc[t] = wmma4(a, glbB(B1p, k, c0 + 16 * t, lane), acc[t]);
    }
    for (int k = 0; k < NU; k += 4) {
      const v2f a = ldsA(u_s, rv, k, lane, SU);
#pragma unroll
      for (int t = 0; t < 8; ++t)
        acc[t] = wmma4(a, glbB(B2p, k, c0 + 16 * t, lane), acc[t]);
    }
#pragma unroll
    for (int t = 0; t < 8; ++t) {
      const int col = c0 + 16 * t + (lane & 15);
      const float bxv = bx[col];
#pragma unroll
      for (int v = 0; v < 8; ++v) {
        const int row = rv + v + ((lane >> 4) << 3);
        out[(size_t)(r0 + row) * NX + col] = acc[t][v] + bxv;
      }
    }
  }
}

// ---------------------------------------------------------------------------
// Launch
// ---------------------------------------------------------------------------
constexpr size_t MAIN_SMEM = (size_t)(3 * MT * SP + MT * SU + 16 * 17 + MT * 16) * 4;
constexpr size_t GJ_SMEM   = (size_t)(256 * 257 + 256) * 4;

extern "C" void kernel_launch(void* const* d_in, const int* in_sizes, int n_in,
                              void* d_out, int out_size, void* d_ws,
                              size_t ws_size, hipStream_t stream) {
  (void)in_sizes; (void)n_in; (void)out_size; (void)ws_size;
  // setup_inputs order: t, xi, u, Pstar, Chi, Y1, B2, D12, X, bx, bv
  const float* xi    = (const float*)d_in[1];
  const float* u     = (const float*)d_in[2];
  const float* Pstar = (const float*)d_in[3];
  const float* Chi   = (const float*)d_in[4];
  const float* Y1    = (const float*)d_in[5];
  const float* B2    = (const float*)d_in[6];
  const float* D12   = (const float*)d_in[7];
  const float* X     = (const float*)d_in[8];
  const float* bx    = (const float*)d_in[9];
  const float* bv    = (const float*)d_in[10];
  float* out = (float*)d_out;

  float* ws   = (float*)d_ws;          // ~3.2 MB of f32 scratch
  float* H    = ws;                    // 512*512
  float* P    = H + 512 * 512;         // 256*256 each below
  float* Pinv = P + 65536;
  float* Y    = Pinv + 65536;
  float* Z    = Y + 65536;
  float* D11p = Z + 65536;             // fragment-packed weights from here on
  float* C1p  = D11p + 65536;
  float* ATp  = C1p + 65536;
  float* B1p  = ATp + 65536;
  float* D12p = B1p + 65536;           // 64*256
  float* B2p  = D12p + 64 * 256;       // 64*256

  hipFuncSetAttribute((const void*)gj_inverse,
                      hipFuncAttributeMaxDynamicSharedMemorySize, (int)GJ_SMEM);
  hipFuncSetAttribute((const void*)rren_main,
                      hipFuncAttributeMaxDynamicSharedMemorySize, (int)MAIN_SMEM);

  syrk_eps<<<dim3(32, 32), 256, 0, stream>>>(X, H, 512, 512, 1.0f, EPSC);
  syrk_eps<<<dim3(16, 16), 256, 0, stream>>>(Pstar, P, 256, 256, 0.5f, EPSC);
  derive<<<dim3(16, 16), 256, 0, stream>>>(H, Chi, Y1, D12, B2, Y, Z, D11p,
                                           C1p, D12p, B2p);
  gj_inverse<<<1, 1024, GJ_SMEM, stream>>>(P, Pinv);
  gemm_tn<<<dim3(16, 16), 256, 0, stream>>>(Y, Pinv, ATp);   // ATp = pack(Y^T Pinv)
  gemm_tn<<<dim3(16, 16), 256, 0, stream>>>(Z, Pinv, B1p);   // B1p = pack(Z^T Pinv)
  rren_main<<<32768 / MT, 256, MAIN_SMEM, stream>>>(xi, u, C1p, D12p, D11p, ATp,
                                                    B1p, B2p, bx, bv, out);
}